// ChebGCNX_15839839387779
// MI455X (gfx1250) — compile-verified
//
#include <hip/hip_runtime.h>

typedef __attribute__((ext_vector_type(2))) float v2f;
typedef __attribute__((ext_vector_type(8))) float v8f;

#define N_NODES 100000
#define E_EDGES 1600000
#define F_DIM   64

// ---------------------------------------------------------------- utilities
__global__ void zero_f32(float* __restrict__ p, int n) {
  int i = blockIdx.x * blockDim.x + threadIdx.x;
  if (i < n) p[i] = 0.0f;
}

__global__ void deg_accum(const int* __restrict__ row, float* __restrict__ deg, int e) {
  int i = blockIdx.x * blockDim.x + threadIdx.x;
  if (i < e) atomicAdd(&deg[row[i]], 1.0f);
}

__global__ void make_dis(float* __restrict__ deg, int n) {
  int i = blockIdx.x * blockDim.x + threadIdx.x;
  if (i < n) {
    float d = deg[i];
    deg[i] = (d > 0.0f) ? rsqrtf(d) : 0.0f;
  }
}

__global__ void edge_weight(const int* __restrict__ row, const int* __restrict__ col,
                            const float* __restrict__ dis, float* __restrict__ w, int e) {
  int i = blockIdx.x * blockDim.x + threadIdx.x;
  if (i < e) w[i] = -dis[row[i]] * dis[col[i]];
}

// Sparse propagation: P[row[e], :] += w[e] * H[col[e], :]
// 16 threads per edge, 4 features each (float4 gather from L2-resident H,
// 4 native GLOBAL_ATOMIC_ADD_F32 into P).
__global__ void scatter_prop(const int* __restrict__ row, const int* __restrict__ col,
                             const float* __restrict__ w, const float* __restrict__ H,
                             float* __restrict__ P, int e) {
  int gid = blockIdx.x * blockDim.x + threadIdx.x;
  int eidx = gid >> 4;
  if (eidx >= e) return;
  int k = (gid & 15) * 4;
  int r = row[eidx];
  int c = col[eidx];
  float we = w[eidx];
  const float4 v = *(const float4*)(H + (size_t)c * F_DIM + k);
  float* dst = P + (size_t)r * F_DIM + k;
  atomicAdd(dst + 0, we * v.x);
  atomicAdd(dst + 1, we * v.y);
  atomicAdd(dst + 2, we * v.z);
  atomicAdd(dst + 3, we * v.w);
}

// ---------------------------------------------------------------- WMMA GEMM
// OUT[n, FOUT] = act( X @ W0 + P @ W1 + b )  with X,P: [n,64], W: [64,FOUT]
// One wave computes a 16-row strip; V_WMMA_F32_16X16X4_F32, K unrolled by 4.
// Weights for BOTH Chebyshev terms are staged once per block into LDS in a
// lane-matched fused layout, so the inner loop is ds_load_b128 + 2 WMMA.
//
// f32 WMMA VGPR layouts (ISA 7.12.2):
//  A (16x4):  lane l -> M = l&15;  K pair = 2*(l>>4) .. +1
//  B (4x16):  lane l -> N = l&15;  K pair = 2*(l>>4) .. +1
//  C/D(16x16): VGPR r -> M = r + 8*(l>>4); N = l&15
template <int FOUT, bool RELU>
__global__ void cheb_gemm(const float* __restrict__ X, const float* __restrict__ P,
                          const float* __restrict__ W0, const float* __restrict__ W1,
                          const float* __restrict__ B, float* __restrict__ OUT) {
  constexpr int NT   = FOUT / 16;        // n-tiles
  constexpr int NKG  = F_DIM / 4;        // 16 k-groups
  constexpr int NENT = NKG * NT * 32;    // staged float4 entries

  // wlds[(kg*NT + t)*32 + lane] = {W0[k][n], W0[k+1][n], W1[k][n], W1[k+1][n]}
  //   with n = t*16 + (lane&15), k = kg*4 + 2*(lane>>4)
  __shared__ float4 wlds[NENT];

  for (int idx = threadIdx.x; idx < NENT; idx += blockDim.x) {
    const int l   = idx & 31;
    const int rem = idx >> 5;
    const int t   = rem % NT;
    const int kg  = rem / NT;
    const int n   = t * 16 + (l & 15);
    const int k   = kg * 4 + ((l >> 4) * 2);
    wlds[idx] = make_float4(W0[(size_t)k * FOUT + n], W0[(size_t)(k + 1) * FOUT + n],
                            W1[(size_t)k * FOUT + n], W1[(size_t)(k + 1) * FOUT + n]);
  }
  __syncthreads();

  const int lane  = threadIdx.x & 31;
  const int wave  = threadIdx.x >> 5;
  const int tile  = blockIdx.x * 4 + wave;   // 16-row tile index
  const int ntile = N_NODES / 16;            // 6250 (exact)
  if (tile >= ntile) return;                 // wave-uniform: EXEC stays all-1s

  const int row0 = tile * 16;
  const int m    = lane & 15;
  const int kk   = (lane >> 4) * 2;

  v8f acc[NT];
#pragma unroll
  for (int t = 0; t < NT; ++t) acc[t] = (v8f)0.0f;

  const float* xrow = X + (size_t)(row0 + m) * F_DIM;
  const float* prow = P + (size_t)(row0 + m) * F_DIM;

#pragma unroll
  for (int kg = 0; kg < NKG; ++kg) {
    const int k0 = kg * 4;
    const v2f aX = *(const v2f*)(xrow + k0 + kk);   // X[m, k0+kk .. +1]
    const v2f aP = *(const v2f*)(prow + k0 + kk);   // P[m, k0+kk .. +1]
#pragma unroll
    for (int t = 0; t < NT; ++t) {
      const float4 wv = wlds[(kg * NT + t) * 32 + lane];  // ds_load_b128
      v2f b0, b1;
      b0.x = wv.x; b0.y = wv.y;   // W0 fragment
      b1.x = wv.z; b1.y = wv.w;   // W1 fragment
      acc[t] = __builtin_amdgcn_wmma_f32_16x16x4_f32(
          false, aX, false, b0, (short)0, acc[t], false, false);
      acc[t] = __builtin_amdgcn_wmma_f32_16x16x4_f32(
          false, aP, false, b1, (short)0, acc[t], false, false);
    }
  }

  const int roff = (lane >> 4) * 8;            // M offset for C/D layout
#pragma unroll
  for (int t = 0; t < NT; ++t) {
    const float bias = B[t * 16 + m];
#pragma unroll
    for (int r = 0; r < 8; ++r) {
      float v = acc[t][r] + bias;
      if (RELU) v = fmaxf(v, 0.0f);
      OUT[(size_t)(row0 + roff + r) * FOUT + t * 16 + m] = v;
    }
  }
}

// ---------------------------------------------------------------- driver
extern "C" void kernel_launch(void* const* d_in, const int* in_sizes, int n_in,
                              void* d_out, int out_size, void* d_ws, size_t ws_size,
                              hipStream_t stream) {
  const float* x    = (const float*)d_in[0];
  const int*   adj  = (const int*)d_in[1];
  const float* W1_0 = (const float*)d_in[2];
  const float* W1_1 = (const float*)d_in[3];
  const float* b1   = (const float*)d_in[4];
  const float* Wx_0 = (const float*)d_in[5];
  const float* Wx_1 = (const float*)d_in[6];
  const float* bx   = (const float*)d_in[7];
  const float* W2_0 = (const float*)d_in[8];
  const float* W2_1 = (const float*)d_in[9];
  const float* b2   = (const float*)d_in[10];
  float* out = (float*)d_out;

  const int* row = adj;
  const int* col = adj + E_EDGES;

  // workspace carve-up (all 16B-aligned offsets)
  float* dis = (float*)d_ws;                         // N
  float* w   = dis + N_NODES;                        // E
  float* P   = w + E_EDGES;                          // N*64
  float* H1  = P + (size_t)N_NODES * F_DIM;          // N*64
  float* H2  = H1 + (size_t)N_NODES * F_DIM;         // N*64

  const int TB = 256;
  const int gN   = (N_NODES + TB - 1) / TB;
  const int gE   = (E_EDGES + TB - 1) / TB;
  const int gNF  = (N_NODES * F_DIM + TB - 1) / TB;
  const int gSc  = (E_EDGES * 16 + TB - 1) / TB;
  const int gGem = (N_NODES / 16 + 3) / 4;           // 4 waves (16-row tiles) / block

  // degree -> dis = rsqrt(deg), then per-edge weights
  zero_f32<<<gN, TB, 0, stream>>>(dis, N_NODES);
  deg_accum<<<gE, TB, 0, stream>>>(row, dis, E_EDGES);
  make_dis<<<gN, TB, 0, stream>>>(dis, N_NODES);
  edge_weight<<<gE, TB, 0, stream>>>(row, col, dis, w, E_EDGES);

  // layer 1: H1 = relu(x@W1_0 + prop(x)@W1_1 + b1)
  zero_f32<<<gNF, TB, 0, stream>>>(P, N_NODES * F_DIM);
  scatter_prop<<<gSc, TB, 0, stream>>>(row, col, w, x, P, E_EDGES);
  cheb_gemm<64, true><<<gGem, 128, 0, stream>>>(x, P, W1_0, W1_1, b1, H1);

  // layer 2: H2 = relu(H1@Wx_0 + prop(H1)@Wx_1 + bx)
  zero_f32<<<gNF, TB, 0, stream>>>(P, N_NODES * F_DIM);
  scatter_prop<<<gSc, TB, 0, stream>>>(row, col, w, H1, P, E_EDGES);
  cheb_gemm<64, true><<<gGem, 128, 0, stream>>>(H1, P, Wx_0, Wx_1, bx, H2);

  // layer 3: out = H2@W2_0 + prop(H2)@W2_1 + b2
  zero_f32<<<gNF, TB, 0, stream>>>(P, N_NODES * F_DIM);
  scatter_prop<<<gSc, TB, 0, stream>>>(row, col, w, H2, P, E_EDGES);
  cheb_gemm<16, false><<<gGem, 128, 0, stream>>>(H2, P, W2_0, W2_1, b2, out);
}